// GraphSage_2972117368898
// MI455X (gfx1250) — compile-verified
//
#include <hip/hip_runtime.h>
#include <hip/hip_bf16.h>

typedef __attribute__((ext_vector_type(2))) float v2f;
typedef __attribute__((ext_vector_type(8))) float v8f;

#define FEAT 128        // D_IN == D_HID == 128 (reference)
#define KTOT 256        // concat(mean, self) -> K = 256
#define A_STRIDE 258    // 256 + 2 pad: conflict-free LDS, keeps 8B alignment
#define MTILES 2        // 16-row tiles per block: B fragment reused 2x per load

// ---------------------------------------------------------------------------
// Zero-fill workspace
// ---------------------------------------------------------------------------
__global__ void k_zero(float* __restrict__ p, long long n) {
    long long i = (long long)blockIdx.x * blockDim.x + threadIdx.x;
    long long stride = (long long)gridDim.x * blockDim.x;
    for (; i < n; i += stride) p[i] = 0.0f;
}

// ---------------------------------------------------------------------------
// Edge scatter: one wave32 per edge. Each lane moves float4 (32*4 = 128 feats)
// via global_atomic_add_f32. deg accumulated once (layer 1 only).
// L2-resident: feature table (25.6MB) + msg buffer both fit in 192MB L2.
// ---------------------------------------------------------------------------
__global__ void k_scatter(const float* __restrict__ feat,
                          const int* __restrict__ src,
                          const int* __restrict__ dst,
                          float* __restrict__ msg,
                          float* __restrict__ deg,
                          int E, int addDeg) {
    int gid  = blockIdx.x * blockDim.x + threadIdx.x;
    int edge = gid >> 5;
    int lane = gid & 31;
    if (edge >= E) return;
    int s = src[edge];
    int d = dst[edge];
    const float4 v = *(const float4*)(feat + (size_t)s * FEAT + lane * 4);
    float* m = msg + (size_t)d * FEAT + lane * 4;
    atomicAdd(m + 0, v.x);
    atomicAdd(m + 1, v.y);
    atomicAdd(m + 2, v.z);
    atomicAdd(m + 3, v.w);
    if (addDeg && lane == 0) atomicAdd(deg + d, 1.0f);
}

// ---------------------------------------------------------------------------
// Fused SAGE layer:  out = [msg/deg | self] @ [Wl | Wr]^T + bias  (opt. ReLU)
// Block = MTILES 16-row M tiles x full Dout. A tiles staged in LDS. Each wave
// owns one 16-wide N tile and MTILES accumulators; each B fragment (weights,
// global/L1-resident) feeds MTILES WMMAs. 64 k-steps of
// V_WMMA_F32_16X16X4_F32 per accumulator over K=256.
//
// WMMA f32 16x16x4 operand layout (ISA 7.12.2):
//   A (16x4): lane&15 = M, lane>>4 selects K {0,1} vs {2,3}; 2 VGPRs = K pair
//   B (4x16): mirrored (lane&15 = N, lane>>4 selects K pair)
//   C/D (16x16): 8 VGPRs; lanes 0-15 hold M=r, lanes 16-31 hold M=r+8
// ---------------------------------------------------------------------------
template <bool RELU>
__global__ void k_sage_gemm(const float* __restrict__ msg,
                            const float* __restrict__ deg,
                            const float* __restrict__ self,
                            const float* __restrict__ Wl,
                            const float* __restrict__ Wr,
                            const float* __restrict__ bias,
                            float* __restrict__ out,
                            int N, int Dout) {
    __shared__ float As[MTILES * 16 * A_STRIDE];
    __shared__ float invd[MTILES * 16];

    const int m0  = blockIdx.x * (MTILES * 16);
    const int tid = threadIdx.x;

    if (tid < MTILES * 16) {
        int r = m0 + tid < N ? m0 + tid : N - 1;
        invd[tid] = 1.0f / fmaxf(deg[r], 1.0f);
    }
    __syncthreads();

    // Stage A tiles = [mean | self], (MTILES*16) rows x 256 cols (coalesced)
    for (int idx = tid; idx < MTILES * 16 * KTOT; idx += blockDim.x) {
        int r = idx >> 8;            // KTOT == 256
        int c = idx & 255;
        int row = m0 + r < N ? m0 + r : N - 1;
        float v;
        if (c < FEAT)
            v = msg[(size_t)row * FEAT + c] * invd[r];
        else
            v = self[(size_t)row * FEAT + (c - FEAT)];
        As[r * A_STRIDE + c] = v;
    }
    __syncthreads();

    const int wave  = tid >> 5;
    const int lane  = tid & 31;
    const int n0    = wave * 16;
    const int mr    = lane & 15;     // M (A) / N (B,C) lane index
    const int n     = n0 + mr;
    const int khalf = lane >> 4;     // 0: K={k,k+1}, 1: K={k+2,k+3}

    v8f acc[MTILES];
    const float bv = bias[n];
#pragma unroll
    for (int t = 0; t < MTILES; ++t)
#pragma unroll
        for (int i = 0; i < 8; ++i) acc[t][i] = bv;

#pragma unroll 4
    for (int k = 0; k < KTOT; k += 4) {
        const int klo = k + 2 * khalf;
        const float* wp = (klo < FEAT) ? (Wl + (size_t)n * FEAT + klo)
                                       : (Wr + (size_t)n * FEAT + (klo - FEAT));
        v2f b = *(const v2f*)wp;
#pragma unroll
        for (int t = 0; t < MTILES; ++t) {
            v2f a = *(const v2f*)&As[(t * 16 + mr) * A_STRIDE + klo];
            acc[t] = __builtin_amdgcn_wmma_f32_16x16x4_f32(
                /*neg_a=*/false, a, /*neg_b=*/false, b,
                /*c_mod=*/(short)0, acc[t], /*reuse_a=*/false, /*reuse_b=*/false);
        }
    }

    if (m0 + MTILES * 16 <= N) {
        // Fast path: full tile, straight-line strided stores.
        // Per VGPR r: lanes 0-15 write 64B row chunk (M=r), lanes 16-31 the
        // M=r+8 chunk -> two contiguous 64B segments per store.
#pragma unroll
        for (int t = 0; t < MTILES; ++t) {
            const int rowbase = m0 + t * 16 + khalf * 8;
            float* p = out + (size_t)rowbase * Dout + n;
#pragma unroll
            for (int r = 0; r < 8; ++r) {
                float v = acc[t][r];
                if (RELU) v = fmaxf(v, 0.0f);
                *p = v;
                p += Dout;
            }
        }
    } else {
        // Tail block only (at most one per launch)
#pragma unroll
        for (int t = 0; t < MTILES; ++t) {
            const int rowbase = m0 + t * 16 + khalf * 8;
#pragma unroll
            for (int r = 0; r < 8; ++r) {
                int row = rowbase + r;
                if (row < N) {
                    float v = acc[t][r];
                    if (RELU) v = fmaxf(v, 0.0f);
                    out[(size_t)row * Dout + n] = v;
                }
            }
        }
    }
}

// ---------------------------------------------------------------------------
// Driver
// ---------------------------------------------------------------------------
extern "C" void kernel_launch(void* const* d_in, const int* in_sizes, int n_in,
                              void* d_out, int out_size, void* d_ws, size_t ws_size,
                              hipStream_t stream) {
    const float* x    = (const float*)d_in[0];
    const int*   ei   = (const int*)  d_in[1];
    const float* W1l  = (const float*)d_in[2];
    const float* b1   = (const float*)d_in[3];
    const float* W1r  = (const float*)d_in[4];
    const float* W2l  = (const float*)d_in[5];
    const float* b2   = (const float*)d_in[6];
    const float* W2r  = (const float*)d_in[7];
    float*       out  = (float*)d_out;

    const int DHID = in_sizes[3];            // 128
    const int DOUT = in_sizes[6];            // 64
    const int DIN  = in_sizes[2] / DHID;     // 128
    const int N    = in_sizes[0] / DIN;      // 50000
    const int E    = in_sizes[1] / 2;        // 600000

    const int* src = ei;
    const int* dst = ei + E;

    float* ws  = (float*)d_ws;
    float* msg = ws;                                   // N * 128
    float* deg = msg + (size_t)N * FEAT;               // N
    float* h   = deg + N;                              // N * 128

    const int gblocks = (N + MTILES * 16 - 1) / (MTILES * 16);
    const int sblocks = (E * 32 + 255) / 256;

    // ---- Layer 1 ----
    {
        long long nz = (long long)N * FEAT + N;        // msg + deg
        k_zero<<<4096, 256, 0, stream>>>(msg, nz);

        k_scatter<<<sblocks, 256, 0, stream>>>(x, src, dst, msg, deg, E, 1);

        k_sage_gemm<true><<<gblocks, (DHID / 16) * 32, 0, stream>>>(
            msg, deg, x, W1l, W1r, b1, h, N, DHID);
    }

    // ---- Layer 2 ----
    {
        long long nz = (long long)N * FEAT;            // msg only (deg reused)
        k_zero<<<4096, 256, 0, stream>>>(msg, nz);

        k_scatter<<<sblocks, 256, 0, stream>>>(h, src, dst, msg, deg, E, 0);

        k_sage_gemm<false><<<gblocks, (DOUT / 16) * 32, 0, stream>>>(
            msg, deg, h, W2l, W2r, b2, out, N, DOUT);
    }
}